// GraphConvolution_69561290326166
// MI455X (gfx1250) — compile-verified
//
#include <hip/hip_runtime.h>

typedef __attribute__((ext_vector_type(2))) float v2f;
typedef __attribute__((ext_vector_type(8))) float v8f;

#define D 128  // D_IN == D_OUT == 128

// ---------------------------------------------------------------------------
// Kernel 1: H = F (n_rows x 128) @ W (128 x 128), fp32 WMMA 16x16x4.
// Block = 256 threads = 8 waves. Wave w of block b owns M-tile
// rows [ (b*8+w)*16, +16 ) and computes ALL 128 output columns:
// 8 accumulators (8 x v8f = 64 VGPRs). The A fragment is loaded once per
// k-step and reused by all 8 WMMAs -> features are read exactly once from
// HBM (51.2 MB); W (64 KB) stays hot in L0/WGP$.
// Per ISA 7.12.2 (wave32):
//   A 16x4 f32: lanes 0-15 -> M=lane, VGPR0/1 = K=k+0,k+1
//               lanes 16-31 -> M=lane-16, VGPR0/1 = K=k+2,k+3
//   B 4x16 f32: lanes 0-15 -> N=lane, VGPR0/1 = K=k+0,k+1
//               lanes 16-31 -> N=lane-16, VGPR0/1 = K=k+2,k+3
//   C/D 16x16:  VGPR r: lanes 0-15 -> M=r, N=lane; lanes 16-31 -> M=r+8.
// ---------------------------------------------------------------------------
__global__ __launch_bounds__(256) void gemm_proj(const float* __restrict__ F,
                                                 const float* __restrict__ W,
                                                 float* __restrict__ H,
                                                 int n_rows) {
  const int wave = threadIdx.x >> 5;
  const int lane = threadIdx.x & 31;
  const int half = lane >> 4;     // 0: lanes 0-15, 1: lanes 16-31
  const int l    = lane & 15;
  const int m0   = (blockIdx.x * 8 + wave) * 16;

  // Clamp row index for out-of-range lanes (EXEC must stay all-ones for WMMA).
  int mrow = m0 + l;
  if (mrow >= n_rows) mrow = n_rows - 1;
  const float* arow  = F + (size_t)mrow * D + half * 2;   // contiguous K pair
  const float* wbase = W + (size_t)(half * 2) * D + l;

  v8f acc[8] = {};
#pragma unroll 2
  for (int k = 0; k < D; k += 4) {
    const v2f a = *(const v2f*)(arow + k);        // global_load_b64 (8B aligned)
    const float* wk = wbase + (size_t)k * D;
#pragma unroll
    for (int nt = 0; nt < 8; ++nt) {
      v2f b;
      b.x = wk[nt * 16];                          // W[k+2h+0][nt*16 + l]
      b.y = wk[nt * 16 + D];                      // W[k+2h+1][nt*16 + l]
      acc[nt] = __builtin_amdgcn_wmma_f32_16x16x4_f32(
          /*neg_a=*/false, a, /*neg_b=*/false, b,
          /*c_mod=*/(short)0, acc[nt], /*reuse_a=*/false, /*reuse_b=*/false);
    }
  }

  if (m0 + 16 <= n_rows) {
    // Full tile: uniform, unguarded stores.
#pragma unroll
    for (int r = 0; r < 8; ++r) {
      float* hrow = H + (size_t)(m0 + r + half * 8) * D + l;
#pragma unroll
      for (int nt = 0; nt < 8; ++nt) hrow[nt * 16] = acc[nt][r];
    }
  } else if (m0 < n_rows) {
    // Partial tail tile: predicated stores.
    for (int r = 0; r < 8; ++r) {
      const int m = m0 + r + half * 8;
      if (m < n_rows) {
        float* hrow = H + (size_t)m * D + l;
        for (int nt = 0; nt < 8; ++nt) hrow[nt * 16] = acc[nt][r];
      }
    }
  }
}

// ---------------------------------------------------------------------------
// Kernel 2: out[n][j] = bias[j]   (d_out is poisoned by the harness)
// ---------------------------------------------------------------------------
__global__ __launch_bounds__(256) void init_bias(float* __restrict__ out,
                                                 const float* __restrict__ bias,
                                                 int total) {
  const int i = blockIdx.x * 256 + threadIdx.x;
  if (i < total) out[i] = bias[i & (D - 1)];
}

// ---------------------------------------------------------------------------
// Kernel 3: for each edge e: out[src[e]][:] += val[e] * H[dst[e]][:]
// One wave per edge; lane j handles 4 consecutive floats (float4 = b128 load,
// 4x global_atomic_add_f32). H (51.2 MB) lives in the 192 MB L2, so both the
// gather and the scatter-adds are L2 traffic, not HBM.
// ---------------------------------------------------------------------------
__global__ __launch_bounds__(256) void edge_scatter(const float* __restrict__ H,
                                                    const int* __restrict__ esrc,
                                                    const int* __restrict__ edst,
                                                    const float* __restrict__ eval,
                                                    float* __restrict__ out,
                                                    int n_edges) {
  // Edge index is wave-uniform; assert it so edge metadata loads go scalar.
  const int e = __builtin_amdgcn_readfirstlane(
      (int)((blockIdx.x * 256 + threadIdx.x) >> 5));
  const int lane = threadIdx.x & 31;
  if (e >= n_edges) return;

  const int   s = esrc[e];
  const int   d = edst[e];
  const float v = eval[e];

  const float4 h4 = *((const float4*)(H + (size_t)d * D) + lane);  // b128
  float* op = out + (size_t)s * D + lane * 4;
  unsafeAtomicAdd(op + 0, v * h4.x);
  unsafeAtomicAdd(op + 1, v * h4.y);
  unsafeAtomicAdd(op + 2, v * h4.z);
  unsafeAtomicAdd(op + 3, v * h4.w);
}

// ---------------------------------------------------------------------------
extern "C" void kernel_launch(void* const* d_in, const int* in_sizes, int n_in,
                              void* d_out, int out_size, void* d_ws, size_t ws_size,
                              hipStream_t stream) {
  const float* features = (const float*)d_in[0];   // [N, 128]
  const int*   edge_src = (const int*)d_in[1];     // [E]
  const int*   edge_dst = (const int*)d_in[2];     // [E]
  const float* edge_val = (const float*)d_in[3];   // [E]
  const float* weight   = (const float*)d_in[4];   // [128, 128]
  const float* bias     = (const float*)d_in[5];   // [128]
  float* out = (float*)d_out;
  float* H   = (float*)d_ws;                       // scratch: N*128 floats (51.2 MB)

  const int n_nodes = in_sizes[0] / D;             // 100000
  const int n_edges = in_sizes[1];                 // 3200000

  // 1) H = features @ W  (fp32 WMMA; 128 rows per block = 8 waves * 16 rows)
  const int blocks = (n_nodes + 127) / 128;
  gemm_proj<<<blocks, 256, 0, stream>>>(features, weight, H, n_nodes);

  // 2) out = broadcast(bias)
  const int total = n_nodes * D;
  init_bias<<<(total + 255) / 256, 256, 0, stream>>>(out, bias, total);

  // 3) scatter-add over edges (8 edges per 256-thread block)
  edge_scatter<<<(n_edges + 7) / 8, 256, 0, stream>>>(H, edge_src, edge_dst,
                                                      edge_val, out, n_edges);
}